// MixtureOfHMM_6640019440089
// MI455X (gfx1250) — compile-verified
//
#include <hip/hip_runtime.h>
#include <hip/hip_bf16.h>

// Problem constants (from reference)
#define B_ 16
#define T_ 1024
#define G_ 32000
#define E_ 512
#define M_ 16
#define S_ 64

typedef __attribute__((ext_vector_type(2))) float v2f;
typedef __attribute__((ext_vector_type(8))) float v8f;

// ---------------------------------------------------------------- utilities
__device__ __forceinline__ float wave_max32(float v) {
#pragma unroll
    for (int o = 16; o > 0; o >>= 1) v = fmaxf(v, __shfl_xor(v, o));
    return v;
}
__device__ __forceinline__ float wave_sum32(float v) {
#pragma unroll
    for (int o = 16; o > 0; o >>= 1) v += __shfl_xor(v, o);
    return v;
}

// 64-thread (2-wave) block reductions using a tiny LDS mailbox
__device__ __forceinline__ float blk64_max(float v, float* red2, int tid) {
    v = wave_max32(v);
    if ((tid & 31) == 0) red2[tid >> 5] = v;
    __syncthreads();
    float r = fmaxf(red2[0], red2[1]);
    __syncthreads();
    return r;
}
__device__ __forceinline__ float blk64_sum(float v, float* red2, int tid) {
    v = wave_sum32(v);
    if ((tid & 31) == 0) red2[tid >> 5] = v;
    __syncthreads();
    float r = red2[0] + red2[1];
    __syncthreads();
    return r;
}

// ---------------------------------------------------------------- kernels

// Zero scratch region
__global__ void k_zero(float* p, int n) {
    int i = blockIdx.x * blockDim.x + threadIdx.x;
    if (i < n) p[i] = 0.0f;
}

// Sum of embeddings per batch row: meansum[b][e] = sum_t embed[x[b,t]][e]
// grid (B, 8), block 512 (one thread per embedding column, 128 tokens per block)
__global__ void k_embed_sum(const int* __restrict__ x,
                            const float* __restrict__ emb,
                            float* __restrict__ meansum) {
    const int b = blockIdx.x;
    const int chunk = blockIdx.y;
    const int e = threadIdx.x;          // 0..511
    __shared__ int toks[128];
    const int t0 = chunk * 128;
    if (e < 128) toks[e] = x[b * T_ + t0 + e];
    __syncthreads();
    float acc = 0.0f;
#pragma unroll 4
    for (int i = 0; i < 128; ++i)
        acc += emb[(size_t)toks[i] * E_ + e];
    atomicAdd(&meansum[b * E_ + e], acc);
}

// logits[b,g] = (meansum[b]/T) . W[g] + bias[g]  via V_WMMA_F32_16X16X4_F32.
// One wave per 16-wide g-tile; M dimension == B == 16 exactly.
// grid 250 blocks x 256 threads (8 waves) -> 2000 g-tiles.
__global__ void k_gemm_wmma(const float* __restrict__ meansum,
                            const float* __restrict__ W,
                            const float* __restrict__ bias,
                            float* __restrict__ logits) {
    const int wave  = threadIdx.x >> 5;
    const int lane  = threadIdx.x & 31;
    const int gtile = blockIdx.x * 8 + wave;
    const int g0    = gtile * 16;

    const int half  = lane >> 4;        // 0: K+0/K+1 ; 1: K+2/K+3
    const int idx16 = lane & 15;        // M index (A) / N index (B,C,D)

    const float invT = 1.0f / (float)T_;
    const float* aptr = meansum + idx16 * E_ + half * 2;            // row m
    const float* bptr = W + (size_t)(g0 + idx16) * E_ + half * 2;   // row g

    v8f c = {};
#pragma unroll 4
    for (int k = 0; k < E_; k += 4) {
        // Branch-free look-ahead prefetch; offset wraps within the row so the
        // address is always in-bounds (max byte offset row*2048 + 2040).
        __builtin_prefetch(bptr + ((k + 64) & (E_ - 1)), 0, 0);  // global_prefetch_b8
        v2f a, bb;
        a.x  = aptr[k] * invT;
        a.y  = aptr[k + 1] * invT;
        bb.x = bptr[k];
        bb.y = bptr[k + 1];
        // D = A(16x4,f32) * B(4x16,f32) + C
        c = __builtin_amdgcn_wmma_f32_16x16x4_f32(
                false, a, false, bb, (short)0, c, false, false);
    }

    const float bv = bias[g0 + idx16];
    // C/D layout: VGPR r -> (M=r, lanes 0-15) and (M=r+8, lanes 16-31), N=idx16
#pragma unroll
    for (int r = 0; r < 8; ++r) {
        const int m = r + half * 8;
        logits[(size_t)m * G_ + g0 + idx16] = c[r] + bv;
    }
}

// Per-batch-row: val[b] = mean_t logits[b, x[b,t]] - logsumexp_g logits[b,:]
// 16 blocks x 1024 threads
__global__ void k_rowreduce(const float* __restrict__ logits,
                            const int* __restrict__ x,
                            float* __restrict__ val) {
    const int b = blockIdx.x;
    const int tid = threadIdx.x;
    __shared__ float red[1024];
    const float* row = logits + (size_t)b * G_;

    float mx = -INFINITY;
    for (int g = tid; g < G_; g += 1024) mx = fmaxf(mx, row[g]);
    red[tid] = mx;
    __syncthreads();
    for (int o = 512; o > 0; o >>= 1) {
        if (tid < o) red[tid] = fmaxf(red[tid], red[tid + o]);
        __syncthreads();
    }
    mx = red[0];
    __syncthreads();

    float s = 0.0f;
    for (int g = tid; g < G_; g += 1024) s += __expf(row[g] - mx);
    red[tid] = s;
    __syncthreads();
    for (int o = 512; o > 0; o >>= 1) {
        if (tid < o) red[tid] += red[tid + o];
        __syncthreads();
    }
    const float lse = mx + __logf(red[0]);
    __syncthreads();

    float ts = 0.0f;
    for (int t = tid; t < T_; t += 1024) ts += row[x[b * T_ + t]];
    red[tid] = ts;
    __syncthreads();
    for (int o = 512; o > 0; o >>= 1) {
        if (tid < o) red[tid] += red[tid + o];
        __syncthreads();
    }
    if (tid == 0) val[b] = red[0] * (1.0f / (float)T_) - lse;
}

// Batch-independent HMM recursion in the exp domain with per-step rescale.
// One block per mixture (64 threads); P (softmax over prev state) lives in LDS.
__global__ void k_hmm(const float* __restrict__ init_dist,
                      const float* __restrict__ trans,
                      float* __restrict__ wout) {
    const int m  = blockIdx.x;
    const int sn = threadIdx.x;         // 0..63 : next-state index
    __shared__ float P[S_ * S_];        // P[sp*64+sn] = exp(log_softmax(trans/2))
    __shared__ float u[S_];
    __shared__ float red2[2];

    // Stage transition/2 into LDS (coalesced; thread sn owns column sn)
    for (int i = sn; i < S_ * S_; i += S_)
        P[i] = trans[(size_t)m * S_ * S_ + i] * 0.5f;
    __syncthreads();

    // Column softmax over sp (each thread exclusively owns column sn)
    float cmx = -INFINITY;
#pragma unroll 8
    for (int sp = 0; sp < S_; ++sp) cmx = fmaxf(cmx, P[sp * S_ + sn]);
    float cs = 0.0f;
#pragma unroll 8
    for (int sp = 0; sp < S_; ++sp) cs += __expf(P[sp * S_ + sn] - cmx);
    const float cinv = 1.0f / cs;
#pragma unroll 8
    for (int sp = 0; sp < S_; ++sp)
        P[sp * S_ + sn] = __expf(P[sp * S_ + sn] - cmx) * cinv;

    // u = softmax(init_dist/2) (== exp(log_softmax)), K = 0
    const float a    = init_dist[m * S_ + sn] * 0.5f;
    const float am   = blk64_max(a, red2, sn);
    const float ea   = __expf(a - am);
    const float asum = blk64_sum(ea, red2, sn);
    u[sn] = ea / asum;
    float K = 0.0f;
    __syncthreads();

    // T steps of v = P^T u ; renormalize by max ; K += log(max)
    for (int t = 0; t < T_; ++t) {
        float v = 0.0f;
#pragma unroll 8
        for (int sp = 0; sp < S_; ++sp)
            v = fmaf(P[sp * S_ + sn], u[sp], v);
        const float vm = blk64_max(v, red2, sn);  // barrier inside: u reads done
        u[sn] = v / vm;
        K += __logf(vm);
        __syncthreads();                          // u visible for next step
    }
    wout[m * S_ + sn] = __logf(u[sn]) + K;
}

// out[b] = logsumexp_{m,s}(w/T) + val[b]
__global__ void k_final(const float* __restrict__ wbuf,
                        const float* __restrict__ val,
                        float* __restrict__ out) {
    const int tid = threadIdx.x;        // 1024 == M*S
    __shared__ float red[1024];
    const float a = wbuf[tid] * (1.0f / (float)T_);
    red[tid] = a;
    __syncthreads();
    for (int o = 512; o > 0; o >>= 1) {
        if (tid < o) red[tid] = fmaxf(red[tid], red[tid + o]);
        __syncthreads();
    }
    const float mx = red[0];
    __syncthreads();
    red[tid] = __expf(a - mx);
    __syncthreads();
    for (int o = 512; o > 0; o >>= 1) {
        if (tid < o) red[tid] += red[tid + o];
        __syncthreads();
    }
    const float scalar = mx + __logf(red[0]);
    if (tid < B_) out[tid] = scalar + val[tid];
}

// ---------------------------------------------------------------- launcher
extern "C" void kernel_launch(void* const* d_in, const int* in_sizes, int n_in,
                              void* d_out, int out_size, void* d_ws, size_t ws_size,
                              hipStream_t stream) {
    const int*   x     = (const int*)d_in[0];
    const float* emb   = (const float*)d_in[1];   // [G,E]
    const float* W     = (const float*)d_in[2];   // [G,E]
    const float* bias  = (const float*)d_in[3];   // [G]
    const float* idist = (const float*)d_in[4];   // [1,M,S]
    const float* trans = (const float*)d_in[5];   // [1,M,S,S]
    // d_in[6] (state_vect) contributes 0.0 * sum(...) -> ignored.

    float* ws      = (float*)d_ws;
    float* meansum = ws;                              // B*E   = 8192
    float* logits  = ws + (B_ * E_);                  // B*G   = 512000
    float* val     = logits + (size_t)B_ * G_;        // B     = 16
    float* wbuf    = val + B_;                        // M*S   = 1024
    float* out     = (float*)d_out;                   // [B,1] f32

    k_zero<<<(B_ * E_ + 1023) / 1024, 1024, 0, stream>>>(meansum, B_ * E_);
    k_embed_sum<<<dim3(B_, T_ / 128), E_, 0, stream>>>(x, emb, meansum);
    k_gemm_wmma<<<(G_ / 16) / 8, 256, 0, stream>>>(meansum, W, bias, logits);
    k_rowreduce<<<B_, 1024, 0, stream>>>(logits, x, val);
    k_hmm<<<M_, S_, 0, stream>>>(idist, trans, wbuf);
    k_final<<<1, M_ * S_, 0, stream>>>(wbuf, val, out);
}